// RNNModel_9646496547398
// MI455X (gfx1250) — compile-verified
//
#include <hip/hip_runtime.h>

// ---------------------------------------------------------------------------
// LSTM LM forward for MI455X (gfx1250, wave32, WMMA).
//   B=256, T=160, V=50000, E=256, H=512, C=5
// Phases:
//   0) pack Wx/Wh into pair-interleaved layout (one b64 per B-fragment lane)
//   1) k_embed_xz : xz[b*T+t, 4H] = emb[tokens] @ Wx + b       (f32 WMMA 16x16x4)
//   2) k_step x T : z = xz_t + h @ Wh ; fused gates, c/h update (f32 WMMA 16x16x4)
//   3) k_preds    : preds = out_h @ U + b2                     (N=5, scalar FMA)
// ---------------------------------------------------------------------------

typedef __attribute__((ext_vector_type(2))) float v2f;
typedef __attribute__((ext_vector_type(4))) float v4f;
typedef __attribute__((ext_vector_type(8))) float v8f;

namespace {
constexpr int B    = 256;
constexpr int T    = 160;
constexpr int E    = 256;
constexpr int H    = 512;
constexpr int COUT = 5;
constexpr int G4H  = 4 * H;   // 2048
constexpr int BT   = B * T;   // 40960
}

__device__ __forceinline__ float fsigmoid(float x) {
    return 1.0f / (1.0f + __expf(-x));
}
__device__ __forceinline__ float ftanh(float x) {
    float e = __expf(-2.0f * x);
    return (1.0f - e) / (1.0f + e);
}

// f32 WMMA: D = A(16x4) * B(4x16) + C(16x16)
__device__ __forceinline__ v8f wmma_f32(v2f a, v2f b, v8f c) {
    return __builtin_amdgcn_wmma_f32_16x16x4_f32(
        /*neg_a=*/false, a, /*neg_b=*/false, b,
        /*c_mod=*/(short)0, c, /*reuse_a=*/false, /*reuse_b=*/false);
}

// ---------------------------------------------------------------------------
// Phase 0a: zero init
// ---------------------------------------------------------------------------
__global__ void k_zero(float* __restrict__ p, int n) {
    int i = blockIdx.x * blockDim.x + threadIdx.x;
    if (i < n) p[i] = 0.0f;
}

// ---------------------------------------------------------------------------
// Phase 0b: pack W[K,N] row-major -> Wp[(k/2)*N*2 + n*2 + {0,1}] = W[k..k+1][n]
// so a B-fragment lane (needs rows k+2h, k+2h+1 at one column) loads one b64.
// ---------------------------------------------------------------------------
__global__ void k_pack_pairs(const float* __restrict__ W, float* __restrict__ Wp,
                             int K, int N) {
    int i = blockIdx.x * blockDim.x + threadIdx.x;      // pair-row * N + col
    if (i >= (K / 2) * N) return;
    int k2 = i / N, n = i % N;
    v2f v;
    v.x = W[(size_t)(2 * k2)     * N + n];
    v.y = W[(size_t)(2 * k2 + 1) * N + n];
    *(v2f*)(Wp + (size_t)i * 2) = v;
}

// ---------------------------------------------------------------------------
// Phase 1: xz[bt, 4H] = emb[tokens[bt]] @ Wx + bias
// Wave = one 16(M) x 64(N) strip: 4 accumulators share one A fragment.
// ---------------------------------------------------------------------------
__global__ __launch_bounds__(256) void k_embed_xz(
    const int*   __restrict__ tokens,   // [BT]
    const float* __restrict__ emb,      // [V, E]
    const float* __restrict__ Wxp,      // [E/2, 4H, 2] pair-packed
    const float* __restrict__ bias,     // [4H]
    float*       __restrict__ xz)       // [BT, 4H]
{
    const int lane  = threadIdx.x & 31;
    const int wid   = threadIdx.x >> 5;
    const int gwave = blockIdx.x * 8 + wid;
    const int NQ    = G4H / 64;                 // 32 N-quads
    const int m0    = (gwave / NQ) * 16;        // row tile base (bt)
    const int nq    = gwave % NQ;               // 64-wide N quad
    const int half  = lane >> 4;
    const int r     = lane & 15;

    const int tok = tokens[m0 + r];             // lanes r and r+16 share row r
    const float* __restrict__ arow = emb + (size_t)tok * E;

    v8f acc[4];
    #pragma unroll
    for (int g = 0; g < 4; ++g) {
        const float bv = bias[nq * 64 + g * 16 + r];
        #pragma unroll
        for (int j = 0; j < 8; ++j) acc[g][j] = bv;
    }

    for (int k = 0; k < E; k += 4) {
        // A frag: lane r+16*half holds A[r][k+2h], A[r][k+2h+1]
        const v2f a = *(const v2f*)(arow + k + 2 * half);
        const size_t prow = (size_t)((k >> 1) + half);   // packed pair-row
        #pragma unroll
        for (int g = 0; g < 4; ++g) {
            const int ncol = nq * 64 + g * 16 + r;
            const v2f bf = *(const v2f*)(Wxp + (prow * G4H + ncol) * 2);
            acc[g] = wmma_f32(a, bf, acc[g]);
        }
    }

    #pragma unroll
    for (int g = 0; g < 4; ++g) {
        const int ncol = nq * 64 + g * 16 + r;
        #pragma unroll
        for (int j = 0; j < 8; ++j) {
            const int m = m0 + j + 8 * half;    // C/D: M = j + 8*half, N = r
            xz[(size_t)m * G4H + ncol] = acc[g][j];
        }
    }
}

// ---------------------------------------------------------------------------
// Phase 2 (xT): one timestep. Wave = 16(batch) x 16(hidden) tile computing
// all 4 gate tiles (i,j,f,o) so the cell update fuses in-register.
// Grid: 16 m-tiles * 32 n-tiles = 512 waves = 64 blocks.
// ---------------------------------------------------------------------------
__global__ __launch_bounds__(256) void k_step(
    const float* __restrict__ xz,      // [B*T, 4H] rows indexed bt = b*T + t
    const float* __restrict__ Whp,     // [H/2, 4H, 2] pair-packed
    const float* __restrict__ h_prev,  // [B, H]
    float*       __restrict__ h_next,  // [B, H]
    float*       __restrict__ cst,     // [B, H] cell state (in-place)
    float*       __restrict__ out_h,   // [B*T, H]
    int t)
{
    const int lane  = threadIdx.x & 31;
    const int wid   = threadIdx.x >> 5;
    const int gwave = blockIdx.x * 8 + wid;     // 0..511
    const int m0    = (gwave >> 5) * 16;        // batch tile
    const int n0    = (gwave & 31) * 16;        // hidden tile
    const int half  = lane >> 4;
    const int r     = lane & 15;

    // init accumulators with precomputed xz (already includes bias)
    v8f acc[4];
    #pragma unroll
    for (int g = 0; g < 4; ++g) {
        #pragma unroll
        for (int j = 0; j < 8; ++j) {
            const int m = m0 + j + 8 * half;                 // batch index
            acc[g][j] = xz[((size_t)m * T + t) * G4H + g * H + n0 + r];
        }
    }

    for (int k = 0; k < H; k += 4) {
        const v2f a = *(const v2f*)(h_prev + (size_t)(m0 + r) * H + k + 2 * half);
        const size_t prow = (size_t)((k >> 1) + half);
        #pragma unroll
        for (int g = 0; g < 4; ++g) {
            const int ncol = g * H + n0 + r;
            const v2f bf = *(const v2f*)(Whp + (prow * G4H + ncol) * 2);
            acc[g] = wmma_f32(a, bf, acc[g]);
        }
    }

    // fused TF-LSTMCell gate math (gate order i, j, f, o; forget_bias = 1.0)
    #pragma unroll
    for (int j = 0; j < 8; ++j) {
        const int m    = m0 + j + 8 * half;
        const int hidx = n0 + r;
        const size_t idx = (size_t)m * H + hidx;
        const float zi = acc[0][j];
        const float zj = acc[1][j];
        const float zf = acc[2][j];
        const float zo = acc[3][j];
        const float cold = cst[idx];
        const float cnew = fsigmoid(zf + 1.0f) * cold + fsigmoid(zi) * ftanh(zj);
        const float hnew = fsigmoid(zo) * ftanh(cnew);
        cst[idx]    = cnew;
        h_next[idx] = hnew;
        out_h[((size_t)m * T + t) * H + hidx] = hnew;
    }
}

// ---------------------------------------------------------------------------
// Phase 3: preds[bt, 5] = out_h[bt, :] @ U + b2.  N=5 -> scalar FMA per
// thread, U staged in LDS, row read vectorized (float4).
// ---------------------------------------------------------------------------
__global__ __launch_bounds__(256) void k_preds(
    const float* __restrict__ out_h,   // [BT, H]
    const float* __restrict__ U,       // [H, 5]
    const float* __restrict__ b2,      // [5]
    float*       __restrict__ preds)   // [BT, 5]
{
    __shared__ float sU[H * COUT];     // 10 KB
    for (int i = threadIdx.x; i < H * COUT; i += blockDim.x) sU[i] = U[i];
    __syncthreads();

    const int bt = blockIdx.x * blockDim.x + threadIdx.x;
    if (bt >= BT) return;

    const float* __restrict__ row = out_h + (size_t)bt * H;
    float acc[COUT];
    #pragma unroll
    for (int c = 0; c < COUT; ++c) acc[c] = b2[c];

    for (int k = 0; k < H; k += 4) {
        const v4f x = *(const v4f*)(row + k);
        #pragma unroll
        for (int kk = 0; kk < 4; ++kk) {
            #pragma unroll
            for (int c = 0; c < COUT; ++c)
                acc[c] = fmaf(x[kk], sU[(k + kk) * COUT + c], acc[c]);
        }
    }
    #pragma unroll
    for (int c = 0; c < COUT; ++c) preds[(size_t)bt * COUT + c] = acc[c];
}

// ---------------------------------------------------------------------------
// Launch
// ---------------------------------------------------------------------------
extern "C" void kernel_launch(void* const* d_in, const int* in_sizes, int n_in,
                              void* d_out, int out_size, void* d_ws, size_t ws_size,
                              hipStream_t stream) {
    const int*   tokens = (const int*)  d_in[0];   // [B,T,1]
    const float* emb    = (const float*)d_in[1];   // [V,E]
    const float* Wx     = (const float*)d_in[2];   // [E,4H]
    const float* Wh     = (const float*)d_in[3];   // [H,4H]
    const float* bias   = (const float*)d_in[4];   // [4H]
    const float* U      = (const float*)d_in[5];   // [H,5]
    const float* b2     = (const float*)d_in[6];   // [5]
    float* preds = (float*)d_out;

    // workspace layout (floats)
    float* xz    = (float*)d_ws;                          // BT*4H   (336 MB)
    float* out_h = xz    + (size_t)BT * G4H;              // BT*H    ( 84 MB)
    float* h0    = out_h + (size_t)BT * H;                // B*H
    float* h1    = h0    + (size_t)B * H;                 // B*H
    float* cst   = h1    + (size_t)B * H;                 // B*H
    float* Wxp   = cst   + (size_t)B * H;                 // E*4H    (  2 MB)
    float* Whp   = Wxp   + (size_t)E * G4H;               // H*4H    (  4 MB)

    // init h0, c = 0
    k_zero<<<(B * H + 255) / 256, 256, 0, stream>>>(h0, B * H);
    k_zero<<<(B * H + 255) / 256, 256, 0, stream>>>(cst, B * H);

    // pack weights (pair-interleave rows) — stays L2-resident afterwards
    {
        int nx = (E / 2) * G4H;
        int nh = (H / 2) * G4H;
        k_pack_pairs<<<(nx + 255) / 256, 256, 0, stream>>>(Wx, Wxp, E, G4H);
        k_pack_pairs<<<(nh + 255) / 256, 256, 0, stream>>>(Wh, Whp, H, G4H);
    }

    // xz = emb[tokens] @ Wx + bias
    const int embed_blocks = (BT / 16) * (G4H / 64) / 8;  // 10240
    k_embed_xz<<<embed_blocks, 256, 0, stream>>>(tokens, emb, Wxp, bias, xz);

    // recurrence: kernel-launch boundary = grid-wide sync per timestep
    for (int t = 0; t < T; ++t) {
        const float* hp = (t & 1) ? h1 : h0;
        float*       hn = (t & 1) ? h0 : h1;
        k_step<<<64, 256, 0, stream>>>(xz, Whp, hp, hn, cst, out_h, t);
    }

    // final projection
    k_preds<<<(BT + 255) / 256, 256, 0, stream>>>(out_h, U, b2, preds);
}